// GCN3_14061722927349
// MI455X (gfx1250) — compile-verified
//
#include <hip/hip_runtime.h>
#include <cstdint>

typedef float v2f __attribute__((ext_vector_type(2)));
typedef float v8f __attribute__((ext_vector_type(8)));

#define DHID 128

// ---------- degree / normalization ----------
__global__ __launch_bounds__(256) void k_zero(float* __restrict__ p, int n) {
    int i = blockIdx.x * blockDim.x + threadIdx.x;
    if (i < n) p[i] = 0.0f;
}

__global__ __launch_bounds__(256) void k_deg(const int* __restrict__ ei,
                                             float* __restrict__ deg, int E) {
    int e = blockIdx.x * blockDim.x + threadIdx.x;
    if (e < E) atomicAdd(&deg[ei[E + e]], 1.0f);   // col = destination
}

__global__ __launch_bounds__(256) void k_dinv(float* __restrict__ d, int N) {
    int i = blockIdx.x * blockDim.x + threadIdx.x;
    if (i < N) d[i] = rsqrtf(d[i] + 1.0f);         // +1 self-loop, deg >= 1
}

// ---------- dense transform: out[N x DOUT] = relu?(H)[N x 128] @ W[128 x DOUT] ----------
// 256 threads = 8 waves. Block stages a 16x128 strip of H AND the whole W matrix in
// LDS. A-tile stride 132 and W stride DOUT+16 make all WMMA-fragment LDS reads
// bank-conflict-free (rows k,k+2 differ by 32 banks -> the two lane-halves hit
// disjoint 16-bank windows). DOUT is a template constant, so the unrolled K-loop
// has pure constant-offset ds_load + v_wmma, no address arithmetic.
template <int DOUT>
__global__ __launch_bounds__(256) void k_gemm_wmma(const float* __restrict__ H,
                                                   const float* __restrict__ W,
                                                   float* __restrict__ out,
                                                   int N, int relu_in) {
    constexpr int WS = DOUT + 16;               // padded W row stride (floats)
    __shared__ float As[16][132];
    __shared__ float Ws[DHID * WS];
    const int r0  = blockIdx.x * 16;
    const int tid = threadIdx.x;

    // Stage 16x128 tile of H into LDS (float4, coalesced), ReLU folded in.
#pragma unroll
    for (int i = 0; i < 2; ++i) {
        int f   = tid + i * 256;                // float4 index 0..511
        int row = f >> 5;                       // 32 float4 per row
        int c4  = f & 31;
        float4 v = make_float4(0.f, 0.f, 0.f, 0.f);
        if (r0 + row < N)
            v = *(const float4*)(H + (size_t)(r0 + row) * DHID + c4 * 4);
        if (relu_in) {
            v.x = fmaxf(v.x, 0.f); v.y = fmaxf(v.y, 0.f);
            v.z = fmaxf(v.z, 0.f); v.w = fmaxf(v.w, 0.f);
        }
        *(float4*)&As[row][c4 * 4] = v;
    }

    // Stage W[128 x DOUT] into LDS with padded stride (float4, coalesced).
    constexpr int C4    = DOUT / 4;             // float4 per W row
    constexpr int NITER = DHID * C4 / 256;      // 16 (D=128) or 8 (D=64)
#pragma unroll
    for (int i = 0; i < NITER; ++i) {
        int f   = tid + i * 256;
        int row = f / C4;
        int c4  = f % C4;
        float4 v = *(const float4*)(W + (size_t)row * DOUT + c4 * 4);
        *(float4*)&Ws[row * WS + c4 * 4] = v;
    }
    __syncthreads();

    const int wave = tid >> 5;
    if (wave * 16 >= DOUT) return;              // whole-wave uniform exit (after barrier)
    const int lane = tid & 31;
    const int rm   = lane & 15;                 // row within tile (A), col within tile (B/C/D)
    const int kh   = lane >> 4;                 // K-half select per ISA A/B layout
    const int col  = wave * 16 + rm;

    v8f acc = {};
#pragma unroll
    for (int kb = 0; kb < DHID; kb += 4) {
        const int k0 = kb + kh * 2;
        v2f a, b;
        a.x = As[rm][k0];
        a.y = As[rm][k0 + 1];
        b.x = Ws[k0 * WS + col];
        b.y = Ws[(k0 + 1) * WS + col];
        acc = __builtin_amdgcn_wmma_f32_16x16x4_f32(
                  /*neg_a=*/false, a, /*neg_b=*/false, b,
                  /*c_mod=*/(short)0, acc, /*reuse_a=*/false, /*reuse_b=*/false);
    }

    // C/D layout: VGPR v -> M = v + kh*8, N = lane&15
#pragma unroll
    for (int v = 0; v < 8; ++v) {
        int r = r0 + v + kh * 8;
        if (r < N) out[(size_t)r * DOUT + col] = acc[v];
    }
}

// ---------- aggregation init: out = bias + dinv^2 * hw (self-loop fused) ----------
template <int DOUT>
__global__ __launch_bounds__(256) void k_agg_init(const float* __restrict__ hw,
                                                  const float* __restrict__ dinv,
                                                  const float* __restrict__ bias,
                                                  float* __restrict__ out, int N) {
    long long gid = (long long)blockIdx.x * blockDim.x + threadIdx.x;
    long long total = (long long)N * DOUT;
    if (gid >= total) return;
    int i = (int)(gid / DOUT);                  // power-of-two -> shift
    int d = (int)(gid & (DOUT - 1));
    float w = dinv[i];
    out[gid] = bias[d] + w * w * hw[gid];
}

// ---------- edge scatter: out[dst] += dinv[src]*dinv[dst] * hw[src] ----------
// One wave handles one edge row for DOUT=128 (32 lanes x float4), half a wave for 64.
template <int DOUT>
__global__ __launch_bounds__(256) void k_agg_edges(const float* __restrict__ hw,
                                                   const int* __restrict__ ei,
                                                   const float* __restrict__ dinv,
                                                   float* __restrict__ out, int E) {
    constexpr int CH = DOUT >> 2;               // float4 chunks per row
    long long gid = (long long)blockIdx.x * blockDim.x + threadIdx.x;
    long long total = (long long)E * CH;
    if (gid >= total) return;
    int e = (int)(gid / CH);                    // power-of-two -> shift
    int c = (int)(gid & (CH - 1));
    int src = ei[e];
    int dst = ei[E + e];
    float w = dinv[src] * dinv[dst];
    const float4 v = *(const float4*)(hw + (size_t)src * DOUT + c * 4);
    float* o = out + (size_t)dst * DOUT + c * 4;
    atomicAdd(o + 0, v.x * w);
    atomicAdd(o + 1, v.y * w);
    atomicAdd(o + 2, v.z * w);
    atomicAdd(o + 3, v.w * w);
}

// ---------- log_softmax over 64 cols, one wave per row, in-place ----------
__global__ __launch_bounds__(256) void k_logsoftmax64(float* __restrict__ io, int N) {
    int row  = (int)((blockIdx.x * (long long)blockDim.x + threadIdx.x) >> 5);
    int lane = threadIdx.x & 31;
    if (row >= N) return;
    float* p = io + (size_t)row * 64;
    float v0 = p[lane];
    float v1 = p[lane + 32];
    float m = fmaxf(v0, v1);
#pragma unroll
    for (int off = 16; off > 0; off >>= 1)
        m = fmaxf(m, __shfl_xor(m, off, 32));
    float s = expf(v0 - m) + expf(v1 - m);
#pragma unroll
    for (int off = 16; off > 0; off >>= 1)
        s += __shfl_xor(s, off, 32);
    float ls = logf(s);
    p[lane]      = v0 - m - ls;
    p[lane + 32] = v1 - m - ls;
}

extern "C" void kernel_launch(void* const* d_in, const int* in_sizes, int n_in,
                              void* d_out, int out_size, void* d_ws, size_t ws_size,
                              hipStream_t stream) {
    const float* x  = (const float*)d_in[0];
    const int*   ei = (const int*)  d_in[1];
    const float* W1 = (const float*)d_in[2];
    const float* b1 = (const float*)d_in[3];
    const float* W2 = (const float*)d_in[4];
    const float* b2 = (const float*)d_in[5];
    const float* W3 = (const float*)d_in[6];
    const float* b3 = (const float*)d_in[7];

    const int N = in_sizes[0] / DHID;   // 100000
    const int E = in_sizes[1] / 2;      // 600000

    float* dinv = (float*)d_ws;
    float* bufA = dinv + (size_t)((N + 255) & ~255);
    float* bufB = bufA + (size_t)N * DHID;
    float* outf = (float*)d_out;

    dim3 b256(256);
    const int gN     = (N + 255) / 256;
    const int gE     = (E + 255) / 256;
    const int gTiles = (N + 15) / 16;
    auto gblk = [](long long t) { return (unsigned)((t + 255) / 256); };

    // normalization coefficients
    k_zero<<<gN, b256, 0, stream>>>(dinv, N);
    k_deg <<<gE, b256, 0, stream>>>(ei, dinv, E);
    k_dinv<<<gN, b256, 0, stream>>>(dinv, N);

    // layer 1: x @ W1 -> agg(+b1); relu deferred into layer-2 GEMM load
    k_gemm_wmma<DHID><<<gTiles, b256, 0, stream>>>(x, W1, bufA, N, 0);
    k_agg_init<DHID> <<<gblk((long long)N * DHID), b256, 0, stream>>>(bufA, dinv, b1, bufB, N);
    k_agg_edges<DHID><<<gblk((long long)E * (DHID / 4)), b256, 0, stream>>>(bufA, ei, dinv, bufB, E);

    // layer 2
    k_gemm_wmma<DHID><<<gTiles, b256, 0, stream>>>(bufB, W2, bufA, N, 1);
    k_agg_init<DHID> <<<gblk((long long)N * DHID), b256, 0, stream>>>(bufA, dinv, b2, bufB, N);
    k_agg_edges<DHID><<<gblk((long long)E * (DHID / 4)), b256, 0, stream>>>(bufA, ei, dinv, bufB, E);

    // layer 3 (DOUT = 64), aggregate straight into d_out
    k_gemm_wmma<64><<<gTiles, b256, 0, stream>>>(bufB, W3, bufA, N, 1);
    k_agg_init<64> <<<gblk((long long)N * 64), b256, 0, stream>>>(bufA, dinv, b3, outf, N);
    k_agg_edges<64><<<gblk((long long)E * 16), b256, 0, stream>>>(bufA, ei, dinv, outf, E);

    // log_softmax in-place (8 waves / block, one row per wave)
    k_logsoftmax64<<<(N + 7) / 8, b256, 0, stream>>>(outf, N);
}